// Encoder_13271448945166
// MI455X (gfx1250) — compile-verified
//
#include <hip/hip_runtime.h>
#include <hip/hip_bf16.h>

typedef float v2f __attribute__((ext_vector_type(2)));
typedef float v8f __attribute__((ext_vector_type(8)));

#define NNODES 10000
#define DIM    256

// ---------------------------------------------------------------------------
// Zero-fill kernel (agg init; bit pattern 0 == float 0, the identity for
// max over post-ReLU (>=0) messages and the isolated-node value).
// ---------------------------------------------------------------------------
__global__ void zero_kernel(float* __restrict__ p, int n) {
    int i = blockIdx.x * blockDim.x + threadIdx.x;
    for (; i < n; i += gridDim.x * blockDim.x) p[i] = 0.0f;
}

// ---------------------------------------------------------------------------
// Tiled GEMM with f32 WMMA:  C = act( A*W^T [+ A2*W2^T] + bias )
//   A:[N,256] row-major, W:[256,256] row-major (we need A @ W^T, so the
//   WMMA B-operand (K x N) is just rows of W read along K).
// Block: 256 thr = 8 waves. Block tile = 16 rows x 128 cols, one 16x16
// WMMA tile per wave. grid = (N/16, 256/128). No divergence -> EXEC all 1s.
// ---------------------------------------------------------------------------
__global__ __launch_bounds__(256)
void gemm_wmma_kernel(const float* __restrict__ A,
                      const float* __restrict__ W,
                      const float* __restrict__ A2,   // optional 2nd product
                      const float* __restrict__ W2,
                      const float* __restrict__ bias,
                      float* __restrict__ C,
                      int do_relu) {
    __shared__ float As[16][DIM + 4];   // +4 pad: row stride 1040B -> distinct banks

    const int t    = threadIdx.x;
    const int lane = t & 31;
    const int wid  = t >> 5;
    const int m16  = lane & 15;          // N index within tile / M row for A
    const int khalf = (lane >> 4) << 1;  // 0 for lanes 0-15, 2 for lanes 16-31

    const int row0 = blockIdx.x * 16;
    const int col0 = blockIdx.y * 128 + wid * 16;

    // Stage A panel (16 x 256) into LDS, 128-bit global loads.
    for (int i = t; i < 16 * (DIM / 4); i += 256) {
        const int r  = i >> 6;
        const int c4 = (i & 63) << 2;
        const float4 v = *(const float4*)(A + (size_t)(row0 + r) * DIM + c4);
        float* dp = &As[r][c4];
        dp[0] = v.x; dp[1] = v.y; dp[2] = v.z; dp[3] = v.w;
    }
    __syncthreads();

    v8f acc = {};
    {
        const float* Wrow = W + (size_t)(col0 + m16) * DIM;
        __builtin_prefetch(Wrow, 0, 3);  // global_prefetch_b8 into L2/L0
        #pragma unroll 8
        for (int k0 = 0; k0 < DIM; k0 += 4) {
            v2f a = *(const v2f*)&As[m16][k0 + khalf];      // ds_load_b64
            v2f b = *(const v2f*)(Wrow + k0 + khalf);       // global_load_b64
            acc = __builtin_amdgcn_wmma_f32_16x16x4_f32(
                false, a, false, b, (short)0, acc, false, false);
        }
    }

    if (A2 != nullptr) {                 // uniform branch: EXEC stays all-1s
        __syncthreads();
        for (int i = t; i < 16 * (DIM / 4); i += 256) {
            const int r  = i >> 6;
            const int c4 = (i & 63) << 2;
            const float4 v = *(const float4*)(A2 + (size_t)(row0 + r) * DIM + c4);
            float* dp = &As[r][c4];
            dp[0] = v.x; dp[1] = v.y; dp[2] = v.z; dp[3] = v.w;
        }
        __syncthreads();
        const float* W2row = W2 + (size_t)(col0 + m16) * DIM;
        __builtin_prefetch(W2row, 0, 3);
        #pragma unroll 8
        for (int k0 = 0; k0 < DIM; k0 += 4) {
            v2f a = *(const v2f*)&As[m16][k0 + khalf];
            v2f b = *(const v2f*)(W2row + k0 + khalf);
            acc = __builtin_amdgcn_wmma_f32_16x16x4_f32(
                false, a, false, b, (short)0, acc, false, false);
        }
    }

    // Epilogue: C/D layout => VGPR r, lanes 0-15: (M=r, N=lane);
    // lanes 16-31: (M=8+r, N=lane-16).
    const int col     = col0 + m16;
    const float bv    = bias[col];
    const int rowbase = row0 + ((lane >> 4) << 3);
    #pragma unroll
    for (int r = 0; r < 8; ++r) {
        float v = acc[r] + bv;
        if (do_relu) v = fmaxf(v, 0.0f);
        C[(size_t)(rowbase + r) * DIM + col] = v;
    }
}

// ---------------------------------------------------------------------------
// Edge scatter-max: agg[dst] = max(agg[dst], m[src]) per feature.
// m >= 0 (post-ReLU) so int-bitwise atomicMax == float max; agg pre-zeroed.
// Block of 256 threads covers 4 edges (64 threads x float4 each).
// ---------------------------------------------------------------------------
__global__ __launch_bounds__(256)
void scatter_max_kernel(const float* __restrict__ m,
                        const int* __restrict__ src,
                        const int* __restrict__ dst,
                        float* __restrict__ agg,
                        int E) {
    const int e = blockIdx.x * 4 + (threadIdx.x >> 6);
    if (e >= E) return;
    const int f = (threadIdx.x & 63) << 2;
    const int s = src[e];
    const int d = dst[e];
    const float4 v = *(const float4*)(m + (size_t)s * DIM + f);
    int* o = (int*)(agg + (size_t)d * DIM + f);
    atomicMax(o + 0, __float_as_int(v.x));
    atomicMax(o + 1, __float_as_int(v.y));
    atomicMax(o + 2, __float_as_int(v.z));
    atomicMax(o + 3, __float_as_int(v.w));
}

// ---------------------------------------------------------------------------
// Fused LayerNorm + ReLU, one 256-thread block per row (d == 256).
// ---------------------------------------------------------------------------
__global__ __launch_bounds__(256)
void ln_relu_kernel(const float* __restrict__ rst,
                    const float* __restrict__ gamma,
                    const float* __restrict__ beta,
                    float* __restrict__ out) {
    __shared__ float red[256];
    const int row = blockIdx.x;
    const int t   = threadIdx.x;
    const float x = rst[(size_t)row * DIM + t];

    red[t] = x;
    __syncthreads();
    #pragma unroll
    for (int s = 128; s > 0; s >>= 1) {
        if (t < s) red[t] += red[t + s];
        __syncthreads();
    }
    const float mu = red[0] * (1.0f / DIM);
    __syncthreads();

    const float d = x - mu;
    red[t] = d * d;
    __syncthreads();
    #pragma unroll
    for (int s = 128; s > 0; s >>= 1) {
        if (t < s) red[t] += red[t + s];
        __syncthreads();
    }
    const float var = red[0] * (1.0f / DIM);

    const float y = d * rsqrtf(var + 1e-5f) * gamma[t] + beta[t];
    out[(size_t)row * DIM + t] = fmaxf(y, 0.0f);
}

// ---------------------------------------------------------------------------
// Driver: two SAGE-pool layers.
//   per layer: m = relu(h@Wp^T+bp); agg = segmax(m[src] -> dst);
//              rst = h@Ws^T + agg@Wn^T + b; h' = relu(LN(rst)*g+be)
// ---------------------------------------------------------------------------
extern "C" void kernel_launch(void* const* d_in, const int* in_sizes, int n_in,
                              void* d_out, int out_size, void* d_ws, size_t ws_size,
                              hipStream_t stream) {
    const float* h0   = (const float*)d_in[0];
    const int*   esrc = (const int*)d_in[1];
    const int*   edst = (const int*)d_in[2];
    const int E = in_sizes[1];

    // per-layer params: Wp, bp, Ws, Wn, b, g, be  at d_in[3 + 7*li + k]
    const float* P[2][7];
    for (int li = 0; li < 2; ++li)
        for (int k = 0; k < 7; ++k)
            P[li][k] = (const float*)d_in[3 + 7 * li + k];

    const size_t NE = (size_t)NNODES * DIM;
    float* buf0 = (float*)d_ws;       // m, then rst (reused)
    float* buf1 = buf0 + NE;          // agg
    float* buf2 = buf1 + NE;          // layer-1 hidden state

    const dim3 gemm_grid(NNODES / 16, DIM / 128);
    const int  scat_grid = (E + 3) / 4;

    const float* h = h0;
    for (int li = 0; li < 2; ++li) {
        const float *Wp = P[li][0], *bp = P[li][1], *Ws = P[li][2],
                    *Wn = P[li][3], *b  = P[li][4], *g  = P[li][5],
                    *be = P[li][6];
        float* hout = (li == 0) ? buf2 : (float*)d_out;

        // m = relu(h @ Wp^T + bp)
        gemm_wmma_kernel<<<gemm_grid, 256, 0, stream>>>(
            h, Wp, nullptr, nullptr, bp, buf0, 1);
        // agg = 0, then segment max over edges
        zero_kernel<<<2048, 256, 0, stream>>>(buf1, (int)NE);
        scatter_max_kernel<<<scat_grid, 256, 0, stream>>>(
            buf0, esrc, edst, buf1, E);
        // rst = h @ Ws^T + agg @ Wn^T + b   (no relu; LN next)
        gemm_wmma_kernel<<<gemm_grid, 256, 0, stream>>>(
            h, Ws, buf1, Wn, b, buf0, 0);
        // h' = relu(layernorm(rst) * g + be)
        ln_relu_kernel<<<NNODES, 256, 0, stream>>>(buf0, g, be, hout);

        h = hout;
    }
}